// GATLayerWithResidual_15934328668761
// MI455X (gfx1250) — compile-verified
//
#include <hip/hip_runtime.h>
#include <hip/hip_bf16.h>
#include <math.h>

typedef __attribute__((ext_vector_type(2))) float v2f;
typedef __attribute__((ext_vector_type(8))) float v8f;

#define NEG_SLOPE 0.2f
#define LN_EPS 1e-5f

// ---------- order-preserving float<->uint encoding for atomic max ----------
__device__ __forceinline__ unsigned enc_f32(float f) {
    unsigned u = __float_as_uint(f);
    return (u & 0x80000000u) ? ~u : (u | 0x80000000u);
}
__device__ __forceinline__ float dec_f32(unsigned u) {
    unsigned v = (u & 0x80000000u) ? (u ^ 0x80000000u) : ~u;
    return __uint_as_float(v);
}

// ---------------------------------------------------------------------------
// Kernel 0: initialize workspace accumulators (harness does not re-zero ws)
// ---------------------------------------------------------------------------
__global__ __launch_bounds__(256) void init_ws_kernel(unsigned* __restrict__ lmax,
                                                      float* __restrict__ denom,
                                                      float* __restrict__ agg,
                                                      int n) {
    long long t = (long long)blockIdx.x * blockDim.x + threadIdx.x;
    if (t < (long long)n * 4) {
        lmax[t] = 0x007FFFFFu;   // enc_f32(-inf)
        denom[t] = 0.0f;
    }
    if (t < (long long)n * 64) agg[t] = 0.0f;
}

// ---------------------------------------------------------------------------
// WMMA GEMM helper: one wave computes a 16x64 output tile (4 col-tiles of 16)
// from a 16x128 A tile in LDS and a 128x64 weight matrix in global memory.
// Direct __restrict__ global pointers keep the loads on the global path
// (no flat fallback) and stores unconditional.
// ---------------------------------------------------------------------------
__device__ __forceinline__ void gemm_tile_16x64(
    const float (*__restrict__ tileA)[128],
    const float* __restrict__ W, const float* __restrict__ bias,
    float* __restrict__ out, int row0, int l16, int half) {
    v8f acc[4] = {};
#pragma unroll 4
    for (int k = 0; k < 128; k += 4) {
        // A fragment: lanes 0-15 hold A[l16][k..k+1], lanes 16-31 A[l16][k+2..k+3]
        v2f a;
        a.x = tileA[l16][k + 2 * half + 0];
        a.y = tileA[l16][k + 2 * half + 1];
        const float* __restrict__ Wk = W + (size_t)(k + 2 * half) * 64 + l16;
#pragma unroll
        for (int t = 0; t < 4; ++t) {
            v2f b;                 // B row-striped across lanes, same K split as A
            b.x = Wk[t * 16];
            b.y = Wk[64 + t * 16];
            acc[t] = __builtin_amdgcn_wmma_f32_16x16x4_f32(
                false, a, false, b, (short)0, acc[t], false, false);
        }
    }
    // Fused bias + store. C/D layout: VGPR r = rows r (lanes 0-15) / r+8 (16-31).
#pragma unroll
    for (int t = 0; t < 4; ++t) {
        float bv = bias[t * 16 + l16];
#pragma unroll
        for (int r = 0; r < 8; ++r) {
            out[(size_t)(row0 + r + 8 * half) * 64 + t * 16 + l16] = acc[t][r] + bv;
        }
    }
}

// ---------------------------------------------------------------------------
// Kernel 1: node transforms xl = x@Wl+bl, xr = x@Wr+br, res = x@res_W+res_b
// One wave (32 lanes) per 16-row tile; V_WMMA_F32_16X16X4_F32, fp32 accum.
// Requires n % 16 == 0 (N = 50000 = 3125 * 16).
// ---------------------------------------------------------------------------
__global__ __launch_bounds__(32) void node_transform_kernel(
    const float* __restrict__ x,
    const float* __restrict__ Wl, const float* __restrict__ bl,
    const float* __restrict__ Wr, const float* __restrict__ br,
    const float* __restrict__ rW, const float* __restrict__ rb,
    float* __restrict__ xl, float* __restrict__ xr, float* __restrict__ res,
    int n) {
    __shared__ __align__(16) float tileA[16][128];   // 8 KB LDS staging of x rows

    const int lane = threadIdx.x;            // 0..31 (wave32)
    const int row0 = blockIdx.x * 16;

    // Coalesced staging: 512 float4 loads spread across 32 lanes.
#pragma unroll
    for (int i = 0; i < 16; ++i) {
        int idx = i * 32 + lane;             // 0..511 float4 slots
        int r = idx >> 5;                    // row within tile
        int c4 = idx & 31;                   // float4 column
        const float4 v = ((const float4*)(x + (size_t)(row0 + r) * 128))[c4];
        *(float4*)&tileA[r][c4 * 4] = v;
    }
    __syncthreads();

    const int half = lane >> 4;              // K-pair selector (ISA 16x4 layout)
    const int l16 = lane & 15;

    gemm_tile_16x64(tileA, Wl, bl, xl, row0, l16, half);
    gemm_tile_16x64(tileA, Wr, br, xr, row0, l16, half);
    gemm_tile_16x64(tileA, rW, rb, res, row0, l16, half);
}

// ---------------------------------------------------------------------------
// Kernel 2: per-(edge,head) attention logits + scatter max (atomic u32 max)
// e-vector (edge_attr @ We) computed on the fly; We/att staged in LDS.
// 4 adjacent lanes = 4 heads of one edge -> gathers of xl/xr coalesce into
// one 256B row per edge; edge_attr / logits streams are fully coalesced.
// ---------------------------------------------------------------------------
__global__ __launch_bounds__(256) void edge_logits_kernel(
    const int* __restrict__ ei, const float* __restrict__ ea,
    const float* __restrict__ We, const float* __restrict__ att,
    const float* __restrict__ xl, const float* __restrict__ xr,
    float* __restrict__ logits, unsigned* __restrict__ lmax,
    long long E, long long E4) {
    __shared__ float sWe[16 * 64];
    __shared__ float sAtt[64];
    for (int i = threadIdx.x; i < 16 * 64; i += blockDim.x) sWe[i] = We[i];
    for (int i = threadIdx.x; i < 64; i += blockDim.x) sAtt[i] = att[i];
    __syncthreads();

    long long t = (long long)blockIdx.x * blockDim.x + threadIdx.x;
    if (t >= E4) return;
    long long e = t >> 2;
    int h = (int)(t & 3);

    int src = ei[e];
    int dst = ei[E + e];

    float ef[16];
    const float* __restrict__ eav = ea + e * 16;
#pragma unroll
    for (int k = 0; k < 16; ++k) ef[k] = eav[k];

    const float* __restrict__ xls = xl + (size_t)src * 64 + h * 16;
    const float* __restrict__ xrd = xr + (size_t)dst * 64 + h * 16;

    float logit = 0.0f;
#pragma unroll
    for (int o = 0; o < 16; ++o) {
        float ev = 0.0f;
#pragma unroll
        for (int k = 0; k < 16; ++k) ev += ef[k] * sWe[k * 64 + h * 16 + o];
        float m = xls[o] + xrd[o] + ev;
        float s = (m > 0.0f) ? m : NEG_SLOPE * m;   // leaky relu
        logit += s * sAtt[h * 16 + o];
    }
    logits[t] = logit;
    atomicMax(&lmax[(size_t)dst * 4 + h], enc_f32(logit));
}

// ---------------------------------------------------------------------------
// Kernel 3: ex = exp(logit - lmax[dst]); scatter-sum into denom
// ---------------------------------------------------------------------------
__global__ __launch_bounds__(256) void edge_exp_kernel(
    const int* __restrict__ ei, float* __restrict__ exbuf,
    const unsigned* __restrict__ lmax, float* __restrict__ denom,
    long long E, long long E4) {
    long long t = (long long)blockIdx.x * blockDim.x + threadIdx.x;
    if (t >= E4) return;
    long long e = t >> 2;
    int h = (int)(t & 3);
    int dst = ei[E + e];
    float lm = dec_f32(lmax[(size_t)dst * 4 + h]);
    if (!isfinite(lm)) lm = 0.0f;
    float ex = __expf(exbuf[t] - lm);
    exbuf[t] = ex;
    atomicAdd(&denom[(size_t)dst * 4 + h], ex);
}

// ---------------------------------------------------------------------------
// Kernel 4: alpha = ex/denom; scatter alpha * xl[src] into agg (f32 atomics)
// ---------------------------------------------------------------------------
__global__ __launch_bounds__(256) void edge_agg_kernel(
    const int* __restrict__ ei, const float* __restrict__ exbuf,
    const float* __restrict__ denom, const float* __restrict__ xl,
    float* __restrict__ agg, long long E, long long E4) {
    long long t = (long long)blockIdx.x * blockDim.x + threadIdx.x;
    if (t >= E4) return;
    long long e = t >> 2;
    int h = (int)(t & 3);
    int src = ei[e];
    int dst = ei[E + e];
    float alpha = exbuf[t] / denom[(size_t)dst * 4 + h];
    const float* __restrict__ xls = xl + (size_t)src * 64 + h * 16;
    float* __restrict__ ad = agg + (size_t)dst * 64 + h * 16;
#pragma unroll
    for (int o = 0; o < 16; ++o) atomicAdd(&ad[o], alpha * xls[o]);
}

// ---------------------------------------------------------------------------
// Kernel 5: bias + LayerNorm(64) + gamma/beta + residual -> d_out
// One wave32 per node, 2 channels per lane, shuffle reductions.
// ---------------------------------------------------------------------------
__global__ __launch_bounds__(256) void node_finalize_kernel(
    const float* __restrict__ agg, const float* __restrict__ cb,
    const float* __restrict__ gamma, const float* __restrict__ beta,
    const float* __restrict__ res, float* __restrict__ out, int n) {
    int wid = threadIdx.x >> 5;
    int lane = threadIdx.x & 31;
    int node = blockIdx.x * 8 + wid;
    if (node >= n) return;

    const float* __restrict__ a = agg + (size_t)node * 64;
    float v0 = a[lane] + cb[lane];
    float v1 = a[lane + 32] + cb[lane + 32];

    float s = v0 + v1;
#pragma unroll
    for (int m = 16; m > 0; m >>= 1) s += __shfl_xor(s, m, 32);
    float mu = s * (1.0f / 64.0f);

    float d0 = v0 - mu, d1 = v1 - mu;
    float q = d0 * d0 + d1 * d1;
#pragma unroll
    for (int m = 16; m > 0; m >>= 1) q += __shfl_xor(q, m, 32);
    float rstd = rsqrtf(q * (1.0f / 64.0f) + LN_EPS);

    const float* __restrict__ r = res + (size_t)node * 64;
    out[(size_t)node * 64 + lane] = d0 * rstd * gamma[lane] + beta[lane] + r[lane];
    out[(size_t)node * 64 + lane + 32] =
        d1 * rstd * gamma[lane + 32] + beta[lane + 32] + r[lane + 32];
}

// ---------------------------------------------------------------------------
extern "C" void kernel_launch(void* const* d_in, const int* in_sizes, int n_in,
                              void* d_out, int out_size, void* d_ws, size_t ws_size,
                              hipStream_t stream) {
    const float* x   = (const float*)d_in[0];
    const int*   ei  = (const int*)d_in[1];
    const float* ea  = (const float*)d_in[2];
    const float* Wl  = (const float*)d_in[3];
    const float* bl  = (const float*)d_in[4];
    const float* Wr  = (const float*)d_in[5];
    const float* br  = (const float*)d_in[6];
    const float* We  = (const float*)d_in[7];
    const float* att = (const float*)d_in[8];
    const float* cb  = (const float*)d_in[9];
    const float* rW  = (const float*)d_in[10];
    const float* rb  = (const float*)d_in[11];
    const float* gam = (const float*)d_in[12];
    const float* bet = (const float*)d_in[13];

    const int n = in_sizes[0] / 128;
    const long long E = in_sizes[1] / 2;
    const long long E4 = E * 4;

    // workspace layout (floats)
    float* ws   = (float*)d_ws;
    float* xl   = ws;                          // n*64
    float* xr   = xl + (size_t)n * 64;         // n*64
    float* res  = xr + (size_t)n * 64;         // n*64
    float* agg  = res + (size_t)n * 64;        // n*64
    unsigned* lmax = (unsigned*)(agg + (size_t)n * 64);  // n*4
    float* denom = (float*)(lmax + (size_t)n * 4);       // n*4
    float* exbuf = denom + (size_t)n * 4;                // E*4

    {   // init accumulators
        long long tot = (long long)n * 64;
        init_ws_kernel<<<(int)((tot + 255) / 256), 256, 0, stream>>>(lmax, denom, agg, n);
    }
    node_transform_kernel<<<n / 16, 32, 0, stream>>>(
        x, Wl, bl, Wr, br, rW, rb, xl, xr, res, n);

    int gE = (int)((E4 + 255) / 256);
    edge_logits_kernel<<<gE, 256, 0, stream>>>(ei, ea, We, att, xl, xr, exbuf, lmax, E, E4);
    edge_exp_kernel<<<gE, 256, 0, stream>>>(ei, exbuf, lmax, denom, E, E4);
    edge_agg_kernel<<<gE, 256, 0, stream>>>(ei, exbuf, denom, xl, agg, E, E4);

    node_finalize_kernel<<<(n + 7) / 8, 256, 0, stream>>>(
        agg, cb, gam, bet, res, (float*)d_out, n);
}